// Mamba_45165876084736
// MI455X (gfx1250) — compile-verified
//
#include <hip/hip_runtime.h>
#include <math.h>

// Mamba-1 forward for MI455X (gfx1250, wave32, WMMA + async global->LDS).
#define DMODEL   1024
#define DSTATE   16
#define DCONV    4
#define DINNER   2048
#define DTRANK   64
#define BSZ      2
#define SEQLEN   2048
#define MROWS    (BSZ * SEQLEN)          // 4096 rows (b,l)
#define XDBL_W   (DTRANK + 2 * DSTATE)   // 96

typedef float v2f __attribute__((ext_vector_type(2)));
typedef float v8f __attribute__((ext_vector_type(8)));

// Issue one 16-byte async global->LDS copy (tracked by ASYNCcnt, no VGPR
// staging; CDNA5 ISA 15.18.3 GLOBAL_LOAD_ASYNC_TO_LDS_B128, GV mode).
__device__ __forceinline__ void async_b128(uint32_t lds_byte_addr, const float* gptr)
{
    unsigned long long ga = (unsigned long long)(uintptr_t)gptr;
    asm volatile("global_load_async_to_lds_b128 %0, %1, off"
                 :: "v"(lds_byte_addr), "v"(ga) : "memory");
}

// ---------------------------------------------------------------------------
// Generic fp32 WMMA GEMM:  C[M,N] = A[M,K] * W[N,K]^T  (+ optional epilogue)
// 256 threads (8 waves). Block tile 64(M) x 32(N), K in chunks of 32 through
// double-buffered LDS filled with async B128 copies; each wave owns one 16x16
// C tile computed with V_WMMA_F32_16X16X4_F32 (full fp32 fidelity; all GEMM
// stages are HBM-bound at 23.3 TB/s, so fp32 math rate is not the limiter).
// EPI == 0: plain store.  EPI == 1: softplus(acc + bias[n]) (dt projection).
// ---------------------------------------------------------------------------
template <int EPI>
__global__ __launch_bounds__(256) void gemm_wmma_f32(
    const float* __restrict__ A, int lda,
    const float* __restrict__ W, int K,
    float* __restrict__ C, int ldc,
    const float* __restrict__ bias)
{
    // Row stride 36 floats: 16B-aligned rows for B128 LDS writes, and
    // 36*r mod 64 = 4*(9r mod 16) -> 16 consecutive rows hit 16 distinct
    // banks => conflict-free fragment column reads.
    __shared__ float sA[2][64][36];
    __shared__ float sW[2][32][36];

    const int tid  = threadIdx.x;
    const int lane = tid & 31;
    const int wave = tid >> 5;
    const int lo   = lane & 15;   // N col (B/C frag) / M row (A frag)
    const int hi   = lane >> 4;   // K-pair selector / M+8 selector
    const int wm   = wave & 3;    // 4 wave-tiles along M
    const int wn   = wave >> 2;   // 2 wave-tiles along N
    const int m0   = blockIdx.y * 64;
    const int n0   = blockIdx.x * 32;

    // Per-thread async-copy coordinates (float4 granularity).
    const int ra = tid >> 3;             // A rows: ra and ra+32
    const int ca = (tid & 7) << 2;       // column in floats (x4 -> 16B)

    v8f c = {};
    const int nch = K >> 5;  // K chunks of 32

    // Issue all 3 B128 copies for one K-chunk into LDS buffer `buf`.
    auto issue = [&](int k0, int buf) {
        uint32_t lA = (uint32_t)(uintptr_t)&sA[buf][0][0];
        uint32_t lW = (uint32_t)(uintptr_t)&sW[buf][0][0];
        async_b128(lA + (uint32_t)(ra * 36 + ca) * 4u,
                   A + (size_t)(m0 + ra) * lda + k0 + ca);
        async_b128(lA + (uint32_t)((ra + 32) * 36 + ca) * 4u,
                   A + (size_t)(m0 + ra + 32) * lda + k0 + ca);
        async_b128(lW + (uint32_t)(ra * 36 + ca) * 4u,
                   W + (size_t)(n0 + ra) * K + k0 + ca);
    };

    // 2-stage pipeline prologue.
    issue(0, 0);
    if (nch > 1) issue(32, 1);

    for (int ch = 0; ch < nch; ++ch) {
        // Async loads complete in order: <=3 pending means chunk `ch` landed
        // (the next chunk's 3 copies may still be in flight).
        if (ch + 1 < nch) asm volatile("s_wait_asynccnt 0x3" ::: "memory");
        else              asm volatile("s_wait_asynccnt 0x0" ::: "memory");
        __syncthreads();  // all waves' async data visible; prev reads done

        const int buf = ch & 1;
        // 8 x V_WMMA_F32_16X16X4_F32 per K-chunk.
        // A frag (16x4): lanes 0-15 hold K=k,k+1; lanes 16-31 hold K=k+2,k+3.
        // B frag (4x16): B[k][n] = W[n][k] -> symmetric LDS indexing.
        #pragma unroll
        for (int kk = 0; kk < 32; kk += 4) {
            v2f a, b;
            a.x = sA[buf][wm * 16 + lo][kk + 2 * hi];
            a.y = sA[buf][wm * 16 + lo][kk + 2 * hi + 1];
            b.x = sW[buf][wn * 16 + lo][kk + 2 * hi];
            b.y = sW[buf][wn * 16 + lo][kk + 2 * hi + 1];
            c = __builtin_amdgcn_wmma_f32_16x16x4_f32(
                    false, a, false, b, (short)0, c, false, false);
        }
        __syncthreads();  // everyone done reading buf -> safe to refill it
        if (ch + 2 < nch) issue((ch + 2) * 32, buf);
    }

    // Epilogue: C layout = VGPR i -> row (i + 8*hi), lanes 0-15 -> col lo.
    const int n = n0 + wn * 16 + lo;
    #pragma unroll
    for (int i = 0; i < 8; ++i) {
        int m = m0 + wm * 16 + hi * 8 + i;
        float v = c[i];
        if (EPI == 1) {  // delta = softplus(acc + dt_proj_b)
            v += bias[n];
            v = (v > 20.f) ? v : log1pf(__expf(v));
        }
        C[(size_t)m * ldc + n] = v;
    }
}

// ---------------------------------------------------------------------------
// Causal depthwise conv1d (width 4) + bias + SiLU.
// xz rows are (b*L+l), x part = cols [0,2048). Re-reads hit L2 (64MB < 192MB).
// ---------------------------------------------------------------------------
__global__ __launch_bounds__(256) void conv_silu_kernel(
    const float* __restrict__ xz,      // [MROWS][2*DINNER]
    const float* __restrict__ conv_w,  // [DINNER][4]
    const float* __restrict__ conv_b,  // [DINNER]
    float* __restrict__ xt)            // [MROWS][DINNER]
{
    int g = blockIdx.x * 256 + threadIdx.x;
    int d = g & (DINNER - 1);
    int m = g >> 11;                    // row (b*L + l)
    int l = m & (SEQLEN - 1);

    float w0 = conv_w[d * 4 + 0], w1 = conv_w[d * 4 + 1];
    float w2 = conv_w[d * 4 + 2], w3 = conv_w[d * 4 + 3];

    float acc = conv_b[d] + w3 * xz[(size_t)m * (2 * DINNER) + d];
    if (l >= 1) acc += w2 * xz[(size_t)(m - 1) * (2 * DINNER) + d];
    if (l >= 2) acc += w1 * xz[(size_t)(m - 2) * (2 * DINNER) + d];
    if (l >= 3) acc += w0 * xz[(size_t)(m - 3) * (2 * DINNER) + d];

    float s = acc * (1.f / (1.f + __expf(-acc)));  // SiLU
    xt[(size_t)m * DINNER + d] = s;
}

// ---------------------------------------------------------------------------
// Selective scan, parallelized over (b, d, n): one lane per state element.
// 16-lane groups share (b,d); y = sum_n h*C via 4 shfl_xor steps (ds path).
// Fused epilogue: y = (y + D[d]*x) * silu(z).
// ---------------------------------------------------------------------------
__global__ __launch_bounds__(256) void scan_kernel(
    const float* __restrict__ xz,     // [MROWS][2*DINNER] (z part at +DINNER)
    const float* __restrict__ xt,     // [MROWS][DINNER]
    const float* __restrict__ delta,  // [MROWS][DINNER]
    const float* __restrict__ xdbl,   // [MROWS][96] (B at +64, C at +80)
    const float* __restrict__ A_log,  // [DINNER][DSTATE]
    const float* __restrict__ Dvec,   // [DINNER]
    float* __restrict__ y)            // [MROWS][DINNER]
{
    int g  = blockIdx.x * 256 + threadIdx.x;
    int n  = g & 15;
    int bd = g >> 4;
    int d  = bd & (DINNER - 1);
    int b  = bd >> 11;

    const float a  = -__expf(A_log[d * DSTATE + n]);  // A = -exp(A_log)
    const float Dd = Dvec[d];
    float h = 0.f;

    const size_t mbase = (size_t)b * SEQLEN;
    for (int l = 0; l < SEQLEN; ++l) {
        size_t m  = mbase + l;
        float dt  = delta[m * DINNER + d];             // broadcast in group
        float xv  = xt[m * DINNER + d];                // broadcast in group
        float Bn  = xdbl[m * XDBL_W + DTRANK + n];     // coalesced 16
        float Cn  = xdbl[m * XDBL_W + DTRANK + DSTATE + n];

        float dA = __expf(dt * a);
        h = dA * h + (dt * xv) * Bn;

        float p = h * Cn;  // reduce 16 state lanes -> y
        p += __shfl_xor(p, 1, 16);
        p += __shfl_xor(p, 2, 16);
        p += __shfl_xor(p, 4, 16);
        p += __shfl_xor(p, 8, 16);

        if (n == 0) {
            float z  = xz[m * (2 * DINNER) + DINNER + d];
            float yo = p + Dd * xv;
            yo *= z * (1.f / (1.f + __expf(-z)));      // * silu(z)
            y[m * DINNER + d] = yo;
        }
    }
}

// ---------------------------------------------------------------------------
extern "C" void kernel_launch(void* const* d_in, const int* in_sizes, int n_in,
                              void* d_out, int out_size, void* d_ws, size_t ws_size,
                              hipStream_t stream)
{
    const float* hid        = (const float*)d_in[0];  // (B, L, 1024)
    const float* in_proj_w  = (const float*)d_in[1];  // (4096, 1024)
    const float* conv_w     = (const float*)d_in[2];  // (2048, 4)
    const float* conv_b     = (const float*)d_in[3];  // (2048,)
    const float* x_proj_w   = (const float*)d_in[4];  // (96, 2048)
    const float* dt_proj_w  = (const float*)d_in[5];  // (2048, 64)
    const float* dt_proj_b  = (const float*)d_in[6];  // (2048,)
    const float* A_log      = (const float*)d_in[7];  // (2048, 16)
    const float* Dv         = (const float*)d_in[8];  // (2048,)
    const float* out_proj_w = (const float*)d_in[9];  // (1024, 2048)
    float* out = (float*)d_out;                       // (B, L, 1024)

    // Workspace layout (fp32): ~170 MB.
    float* ws    = (float*)d_ws;
    float* xz    = ws;                                  // [4096][4096]  64 MB
    float* xt    = xz    + (size_t)MROWS * 2 * DINNER;  // [4096][2048]  32 MB
    float* xdbl  = xt    + (size_t)MROWS * DINNER;      // [4096][96]   1.5 MB
    float* delta = xdbl  + (size_t)MROWS * XDBL_W;      // [4096][2048]  32 MB
    float* yb    = delta + (size_t)MROWS * DINNER;      // [4096][2048]  32 MB

    dim3 blk(256);

    // 1) xz = hidden @ in_proj_w^T           M=4096 K=1024 N=4096
    gemm_wmma_f32<0><<<dim3(4096 / 32, MROWS / 64), blk, 0, stream>>>(
        hid, DMODEL, in_proj_w, DMODEL, xz, 2 * DINNER, nullptr);

    // 2) xt = silu(causal_conv(x) + conv_b)
    conv_silu_kernel<<<(MROWS * DINNER) / 256, blk, 0, stream>>>(
        xz, conv_w, conv_b, xt);

    // 3) x_dbl = xt @ x_proj_w^T             M=4096 K=2048 N=96
    gemm_wmma_f32<0><<<dim3(XDBL_W / 32, MROWS / 64), blk, 0, stream>>>(
        xt, DINNER, x_proj_w, DINNER, xdbl, XDBL_W, nullptr);

    // 4) delta = softplus(x_dbl[:, :64] @ dt_proj_w^T + dt_proj_b)
    //    A is a 96-strided view of x_dbl.    M=4096 K=64 N=2048
    gemm_wmma_f32<1><<<dim3(DINNER / 32, MROWS / 64), blk, 0, stream>>>(
        xdbl, XDBL_W, dt_proj_w, DTRANK, delta, DINNER, dt_proj_b);

    // 5) selective scan (fused +D*x and *silu(z))
    scan_kernel<<<(BSZ * DINNER * DSTATE) / 256, blk, 0, stream>>>(
        xz, xt, delta, xdbl, A_log, Dv, yb);

    // 6) out = y @ out_proj_w^T              M=4096 K=2048 N=1024
    gemm_wmma_f32<0><<<dim3(DMODEL / 32, MROWS / 64), blk, 0, stream>>>(
        yb, DINNER, out_proj_w, DINNER, out, DMODEL, nullptr);
}